// ConvLayer_72310069396093
// MI455X (gfx1250) — compile-verified
//
#include <hip/hip_runtime.h>

typedef __attribute__((ext_vector_type(2))) float v2f;
typedef __attribute__((ext_vector_type(8))) float v8f;

#define N_NODES 50000
#define N_EDGES 200000
#define OUT 32
#define IN_SELF 128
#define EDGE_DIM 16
#define N_TILES (N_NODES / 16)   // 3125, exact

// workspace layout (in floats)
#define WS_WSUM 0          // 512  : Wsum[32][16]
#define WS_BSUM 512        // 32   : bsum[32]
#define WS_DEG  544        // 50000: degree
#define WS_AGG  50544      // 50000*32 : agg sums (8B-aligned: 50544 even)

// ---------------------------------------------------------------------------
// Kernel 0: fold W_edge/b_edge over i, zero deg+agg.
// ---------------------------------------------------------------------------
__global__ void gnn_init_kernel(const float* __restrict__ W_edge,
                                const float* __restrict__ b_edge,
                                float* __restrict__ ws) {
    int tid = blockIdx.x * blockDim.x + threadIdx.x;
    const int totalZero = N_NODES + N_NODES * OUT;
    for (int i = tid; i < totalZero; i += gridDim.x * blockDim.x)
        ws[WS_DEG + i] = 0.0f;
    if (tid < 512) {
        int j = tid >> 4, k = tid & 15;
        float s = 0.0f;
        #pragma unroll
        for (int i = 0; i < 32; ++i)
            s += W_edge[(i * 32 + j) * 16 + k];
        ws[WS_WSUM + tid] = s;
    } else if (tid < 544) {
        int j = tid - 512;
        float s = 0.0f;
        #pragma unroll
        for (int i = 0; i < 32; ++i)
            s += b_edge[i * 32 + j];
        ws[WS_BSUM + j] = s;
    }
}

// ---------------------------------------------------------------------------
// Kernel 1: per-edge message + atomic scatter.
// s[e,j] = h_neigh[src[e],j] * (bsum[j] + ef[e,:]·Wsum[j,:])
// ---------------------------------------------------------------------------
__global__ void gnn_edge_kernel(const float* __restrict__ h_neigh,
                                const float* __restrict__ edge_features,
                                const int* __restrict__ src,
                                const int* __restrict__ dst,
                                float* __restrict__ ws) {
    __shared__ __align__(16) float sW[512];
    __shared__ __align__(16) float sB[32];
    int tid = threadIdx.x;
    if (tid < 512) sW[tid] = ws[WS_WSUM + tid];
    if (tid < 32)  sB[tid] = ws[WS_BSUM + tid];
    __syncthreads();

    int e = blockIdx.x * blockDim.x + tid;
    if (e >= N_EDGES) return;

    int s = src[e];
    int d = dst[e];

    float ef[16];
    const float4* efp = (const float4*)(edge_features + (long long)e * 16);
    #pragma unroll
    for (int q = 0; q < 4; ++q) {
        float4 v = efp[q];
        ef[q * 4 + 0] = v.x; ef[q * 4 + 1] = v.y;
        ef[q * 4 + 2] = v.z; ef[q * 4 + 3] = v.w;
    }

    const float* hn = h_neigh + (long long)s * 32;
    float* aggrow = ws + WS_AGG + (long long)d * 32;

    #pragma unroll
    for (int j = 0; j < 32; ++j) {
        float ew = sB[j];
        #pragma unroll
        for (int k = 0; k < 16; ++k)
            ew = fmaf(ef[k], sW[j * 16 + k], ew);
        atomicAdd(&aggrow[j], hn[j] * ew);
    }
    atomicAdd(&ws[WS_DEG + d], 1.0f);
}

// ---------------------------------------------------------------------------
// Kernel 2: z = relu(h_self @ W_self.T + (agg/deg) @ W_neigh.T) via
// V_WMMA_F32_16X16X4_F32. One wave -> 16x32 output tile (2 accumulators).
// A layout: row = lane&15, K = vgpr + 2*(lane>=16)  -> contiguous float2 load
// B layout: col = lane&15, K = vgpr + 2*(lane>=16)
// C/D: vgpr r holds M = r + 8*(lane>=16), N = lane&15
// ---------------------------------------------------------------------------
__global__ void gnn_node_kernel(const float* __restrict__ h_self,
                                const float* __restrict__ W_self,
                                const float* __restrict__ W_neigh,
                                const float* __restrict__ ws,
                                float* __restrict__ out) {
    __shared__ __align__(16) float sWs[32 * 128];  // W_self, row-major [32][128]
    __shared__ __align__(16) float sWn[32 * 32];   // W_neigh, row-major [32][32]

    int tid = threadIdx.x;
    #pragma unroll
    for (int q = 0; q < 4; ++q)
        ((float4*)sWs)[tid + q * 256] = ((const float4*)W_self)[tid + q * 256];
    ((float4*)sWn)[tid] = ((const float4*)W_neigh)[tid];
    __syncthreads();

    int wave = tid >> 5;
    int lane = tid & 31;
    int tile = blockIdx.x * 8 + wave;
    if (tile >= N_TILES) return;   // wave-uniform: EXEC stays all-1s for WMMA

    int m0   = tile * 16;
    int n    = lane & 15;          // A: row-in-tile ; B: col-in-subtile
    int koff = (lane >> 4) << 1;   // 0 or 2
    int row  = m0 + n;

    v8f acc0 = {};
    v8f acc1 = {};

    // ---- h_self @ W_self.T  (K = 128) ----
    const float* arow = h_self + (long long)row * 128 + koff;
    const float* b0p  = sWs + n * 128 + koff;          // output cols 0..15
    const float* b1p  = sWs + (n + 16) * 128 + koff;   // output cols 16..31
    #pragma unroll
    for (int k = 0; k < 128; k += 4) {
        v2f a  = *(const v2f*)(arow + k);
        v2f b0 = *(const v2f*)(b0p + k);
        v2f b1 = *(const v2f*)(b1p + k);
        acc0 = __builtin_amdgcn_wmma_f32_16x16x4_f32(false, a, false, b0,
                                                     (short)0, acc0, false, false);
        acc1 = __builtin_amdgcn_wmma_f32_16x16x4_f32(false, a, false, b1,
                                                     (short)0, acc1, false, false);
    }

    // ---- (agg/deg) @ W_neigh.T  (K = 32) ----
    float dg   = ws[WS_DEG + row];
    float rdeg = 1.0f / fmaxf(dg, 1.0f);
    const float* grow = ws + WS_AGG + (long long)row * 32 + koff;
    const float* c0p  = sWn + n * 32 + koff;
    const float* c1p  = sWn + (n + 16) * 32 + koff;
    #pragma unroll
    for (int k = 0; k < 32; k += 4) {
        v2f a = *(const v2f*)(grow + k);
        a.x *= rdeg;
        a.y *= rdeg;
        v2f b0 = *(const v2f*)(c0p + k);
        v2f b1 = *(const v2f*)(c1p + k);
        acc0 = __builtin_amdgcn_wmma_f32_16x16x4_f32(false, a, false, b0,
                                                     (short)0, acc0, false, false);
        acc1 = __builtin_amdgcn_wmma_f32_16x16x4_f32(false, a, false, b1,
                                                     (short)0, acc1, false, false);
    }

    // ---- relu + store ----
    int mbase = m0 + ((lane >> 4) << 3);
    #pragma unroll
    for (int r = 0; r < 8; ++r) {
        int m = mbase + r;
        out[m * 32 + n]      = fmaxf(acc0[r], 0.0f);
        out[m * 32 + n + 16] = fmaxf(acc1[r], 0.0f);
    }
}

extern "C" void kernel_launch(void* const* d_in, const int* in_sizes, int n_in,
                              void* d_out, int out_size, void* d_ws, size_t ws_size,
                              hipStream_t stream) {
    const float* h_neigh       = (const float*)d_in[0];
    const float* h_self        = (const float*)d_in[1];
    const float* edge_features = (const float*)d_in[2];
    const float* W_edge        = (const float*)d_in[3];
    const float* b_edge        = (const float*)d_in[4];
    const float* W_self        = (const float*)d_in[5];
    const float* W_neigh       = (const float*)d_in[6];
    const int*   src           = (const int*)d_in[7];
    const int*   dst           = (const int*)d_in[8];
    float*       out           = (float*)d_out;
    float*       ws            = (float*)d_ws;

    const int zeroTotal = N_NODES + N_NODES * OUT;
    gnn_init_kernel<<<(zeroTotal + 255) / 256, 256, 0, stream>>>(W_edge, b_edge, ws);
    gnn_edge_kernel<<<(N_EDGES + 255) / 256, 256, 0, stream>>>(h_neigh, edge_features,
                                                               src, dst, ws);
    gnn_node_kernel<<<(N_TILES + 7) / 8, 256, 0, stream>>>(h_self, W_self, W_neigh,
                                                           ws, out);
}